// GaussianEmbeddingDP_76905684402858
// MI455X (gfx1250) — compile-verified
//
#include <hip/hip_runtime.h>
#include <hip/hip_bf16.h>
#include <stdint.h>

// GaussianEmbeddingDP for MI455X (gfx1250):
//  - one workgroup (256 threads = 8 wave32 waves) per 8192-float row
//  - row staged into LDS once via GLOBAL_LOAD_ASYNC_TO_LDS_B128 (ASYNCcnt),
//    both passes (norm, scale+noise) read LDS -> 1R + 1W of HBM (~46us roofline)
//  - 256 per-thread sum-of-squares partials reduced with V_WMMA_F32_16X16X4_F32
//    (ones B-matrix => row sums), then shfl_xor(16) for the grand total
//  - noise: PCG hash + Box-Muller with fast-math intrinsics (deterministic per element)

#define COLS 8192
#define TPB 256
#define VPT (COLS / 4 / TPB) // float4 vectors per thread = 8

typedef __attribute__((ext_vector_type(2))) float v2f;
typedef __attribute__((ext_vector_type(8))) float v8f;

__device__ __forceinline__ uint32_t pcg_hash(uint32_t v) {
    v = v * 747796405u + 2891336453u;
    uint32_t w = ((v >> ((v >> 28u) + 4u)) ^ v) * 277803737u;
    return (w >> 22u) ^ w;
}

__device__ __forceinline__ void gauss_pair(uint32_t p, float& z0, float& z1) {
    uint32_t h1 = pcg_hash(p * 2u + 0x9E3779B9u);
    uint32_t h2 = pcg_hash(p * 2u + 0x85EBCA6Bu);
    float u1 = ((h1 >> 8) + 1u) * 0x1p-24f; // (0,1]
    float u2 = (h2 >> 8) * 0x1p-24f;        // [0,1)
    float r = __fsqrt_rn(-2.0f * __logf(u1));
    float s, c;
    __sincosf(6.28318530717958647f * u2, &s, &c);
    z0 = r * c;
    z1 = r * s;
}

__global__ __launch_bounds__(TPB) void GaussianEmbeddingDP_76905684402858_kernel(
    const float* __restrict__ in, float* __restrict__ out) {
    __shared__ float smem[COLS];
    __shared__ float part[TPB];
    __shared__ float s_scale;

    const int tid = threadIdx.x;
    const int row = blockIdx.x;
    const uint64_t gbase = (uint64_t)(uintptr_t)(in + (size_t)row * COLS);
    // Flat shared-aperture address: low 32 bits are the LDS byte offset (ISA 10.2)
    const uint32_t lds0 = (uint32_t)(uintptr_t)(&smem[0]);

    // ---- async-DMA the row into LDS (CDNA5 async-to-LDS path, ASYNCcnt) ----
#pragma unroll
    for (int j = 0; j < VPT; ++j) {
        uint32_t idx4 = (uint32_t)(j * TPB + tid);
        uint32_t boff = idx4 * 16u;       // byte offset within row (VADDR, GVS mode)
        uint32_t ldsa = lds0 + boff;      // LDS dest byte address (VDST)
        asm volatile("global_load_async_to_lds_b128 %0, %1, %2"
                     :: "v"(ldsa), "v"(boff), "s"(gbase)
                     : "memory");
    }
    asm volatile("s_wait_asynccnt 0" ::: "memory");
    __syncthreads();

    // ---- pass 1: per-thread partial sum of squares (from LDS) ----
    const float4* s4 = reinterpret_cast<const float4*>(smem);
    float ssq = 0.0f;
#pragma unroll
    for (int j = 0; j < VPT; ++j) {
        float4 v = s4[j * TPB + tid];
        ssq = fmaf(v.x, v.x, ssq);
        ssq = fmaf(v.y, v.y, ssq);
        ssq = fmaf(v.z, v.z, ssq);
        ssq = fmaf(v.w, v.w, ssq);
    }
    part[tid] = ssq;
    __syncthreads();

    // ---- WMMA reduction of the 256 partials (wave 0, EXEC all-ones) ----
    // A[m][k] := part[ch*64 + m*4 + k] (16x4 f32 layout: lanes 0-15 M,K={0,1};
    // lanes 16-31 M,K={2,3}); B = ones(4x16) => D[m][n] += row-sum of A.
    if (tid < 32) {
        const int m = tid & 15;
        const int kh = (tid >> 4) << 1; // 0 or 2
        v2f ones;
        ones[0] = 1.0f;
        ones[1] = 1.0f;
        v8f c = {};
#pragma unroll
        for (int ch = 0; ch < 4; ++ch) {
            const int b = ch * 64 + m * 4 + kh;
            v2f a;
            a[0] = part[b];
            a[1] = part[b + 1];
            c = __builtin_amdgcn_wmma_f32_16x16x4_f32(
                false, a, false, ones, (short)0, c, false, false);
        }
        // lane n: c[0..7] = D[0..7][n]; lane n+16: c[0..7] = D[8..15][n]
        float t = ((c[0] + c[1]) + (c[2] + c[3])) + ((c[4] + c[5]) + (c[6] + c[7]));
        t += __shfl_xor(t, 16, 32); // full 256-element total in every lane
        if (tid == 0) {
            float norm = __fsqrt_rn(t);
            s_scale = (norm > 1.0f) ? (1.0f / norm) : 1.0f; // L2_NORM_CLIP = 1
        }
    }
    __syncthreads();

    // ---- pass 2: scale + Gaussian noise, streaming store ----
    const float sc = s_scale;
    float4* o4 = reinterpret_cast<float4*>(out + (size_t)row * COLS);
    const uint32_t row_e = (uint32_t)row * (uint32_t)COLS;
#pragma unroll
    for (int j = 0; j < VPT; ++j) {
        uint32_t idx4 = (uint32_t)(j * TPB + tid);
        float4 v = s4[idx4];
        uint32_t e = row_e + idx4 * 4u; // even element index
        float z0, z1, z2, z3;
        gauss_pair(e >> 1, z0, z1);
        gauss_pair((e >> 1) + 1u, z2, z3);
        float4 r;
        r.x = fmaf(v.x, sc, 1.1f * z0);
        r.y = fmaf(v.y, sc, 1.1f * z1);
        r.z = fmaf(v.z, sc, 1.1f * z2);
        r.w = fmaf(v.w, sc, 1.1f * z3);
        o4[idx4] = r;
    }
}

extern "C" void kernel_launch(void* const* d_in, const int* in_sizes, int n_in,
                              void* d_out, int out_size, void* d_ws, size_t ws_size,
                              hipStream_t stream) {
    (void)n_in;
    (void)d_ws;
    (void)ws_size;
    const float* in = (const float*)d_in[0];
    float* out = (float*)d_out;
    int rows = (in_sizes[0] > 0) ? (in_sizes[0] / COLS) : (out_size / COLS);
    GaussianEmbeddingDP_76905684402858_kernel<<<rows, TPB, 0, stream>>>(in, out);
}